// BornMayerHugginsForce_50757923504446
// MI455X (gfx1250) — compile-verified
//
#include <hip/hip_runtime.h>

// Born-Mayer-Huggins energy + forces, N=4096, full NxN pair-parameter matrices.
// Memory-bound: 335 MB of params => ~14.4us floor @ 23.3 TB/s. Design:
//  - coords staged to LDS (SoA, 48KB) once per block; b128 broadcast reads
//  - 5 param matrices streamed with 16B (b128) per-lane loads, one full 128B
//    cacheline per stream per outer iteration; explicit near-cache prefetch
//  - per-pair math in exact f32 (v_rsq / v_rcp / v_exp native ops)
//  - Sum_j force accumulation on the matrix pipe: V_WMMA_F32_16X16X4_F32 with
//    an all-ones B (exact f32 row sums, replicated per column -> lanes 0/16
//    hold the finished per-row forces, no shuffle tree needed)

#define NATOMS    4096
#define BOXL      44.0f
#define INV_BOX   (1.0f / 44.0f)
#define CUT2      306.25f       // 17.5^2
#define LN2_INV   1.4426950408889634f

#define ROWS_PER_WAVE   16
#define WAVES_PER_BLOCK 8
#define THREADS         (WAVES_PER_BLOCK * 32)
#define J_PER_WAVE      (NATOMS / WAVES_PER_BLOCK)   // 512
#define OUT_ELEMS       (1 + NATOMS * 3)

typedef float v2f __attribute__((ext_vector_type(2)));
typedef float v4f __attribute__((ext_vector_type(4)));
typedef float v8f __attribute__((ext_vector_type(8)));

__global__ __launch_bounds__(THREADS) void bmh_zero(float* __restrict__ out, int n) {
    int i = blockIdx.x * blockDim.x + threadIdx.x;
    if (i < n) out[i] = 0.0f;
}

__global__ __launch_bounds__(THREADS) void bmh_kernel(
    const float* __restrict__ coords,
    const float* __restrict__ Amat, const float* __restrict__ Cmat,
    const float* __restrict__ Dmat, const float* __restrict__ Rho,
    const float* __restrict__ Sig, float* __restrict__ out)
{
    __shared__ float ldsX[NATOMS];
    __shared__ float ldsY[NATOMS];
    __shared__ float ldsZ[NATOMS];

    for (int j = threadIdx.x; j < NATOMS; j += THREADS) {
        ldsX[j] = coords[3 * j + 0];
        ldsY[j] = coords[3 * j + 1];
        ldsZ[j] = coords[3 * j + 2];
    }
    __syncthreads();

    const int wave = threadIdx.x >> 5;
    const int lane = threadIdx.x & 31;
    const int m    = lane & 15;        // A-matrix row M for this lane
    const int kh   = lane >> 4;        // half-wave: K={0,1} vs K={2,3}
    const int i0   = blockIdx.x * ROWS_PER_WAVE;
    const int i    = i0 + m;
    const int jbeg = wave * J_PER_WAVE;

    const float cxi = ldsX[i];
    const float cyi = ldsY[i];
    const float czi = ldsZ[i];
    const int   ibk  = i0 >> 9;        // 512-block id: uniform across the 16-row group
    const int   iloc = i & 511;

    const size_t row = (size_t)i * NATOMS;
    const float* __restrict__ pA = Amat + row;
    const float* __restrict__ pC = Cmat + row;
    const float* __restrict__ pD = Dmat + row;
    const float* __restrict__ pR = Rho  + row;
    const float* __restrict__ pS = Sig  + row;

    v8f accX = {}; v8f accY = {}; v8f accZ = {};
    const v2f ones = {1.0f, 1.0f};
    float e = 0.0f;   // accumulates Sum(w*pe); *0.5 once at the end

    // outer iter = 32 j-columns = one 128B cacheline per stream
    for (int j0 = jbeg; j0 < jbeg + J_PER_WAVE; j0 += 32) {
        // one unique-line prefetch per stream per outer iter, 4KB ahead
        {
            const int jp = j0 + 4 * kh + 1024;
            __builtin_prefetch(pA + jp, 0, 3);
            __builtin_prefetch(pC + jp, 0, 3);
            __builtin_prefetch(pD + jp, 0, 3);
            __builtin_prefetch(pR + jp, 0, 3);
            __builtin_prefetch(pS + jp, 0, 3);
        }
#pragma unroll
        for (int u = 0; u < 4; ++u) {
            const int jc = j0 + 8 * u;        // 8-column chunk (never crosses 512)
            const int jl = jc + 4 * kh;       // this lane's 4 consecutive columns

            v4f a  = *(const v4f*)(pA + jl);
            v4f cc = *(const v4f*)(pC + jl);
            v4f dd = *(const v4f*)(pD + jl);
            v4f rh = *(const v4f*)(pR + jl);
            v4f sg = *(const v4f*)(pS + jl);
            v4f xj = *(const v4f*)(ldsX + jl);   // half-wave broadcast b128
            v4f yj = *(const v4f*)(ldsY + jl);
            v4f zj = *(const v4f*)(ldsZ + jl);

            // wave-uniform per chunk: same 512-block => weight 1 else 2
            const float w0 = ((jc >> 9) == ibk) ? 1.0f : 2.0f;

            v4f gx, gy, gz;
#pragma unroll
            for (int t = 0; t < 4; ++t) {
                const int j = jl + t;
                float rx = cxi - xj[t]; rx = __builtin_fmaf(-rintf(rx * INV_BOX), BOXL, rx);
                float ry = cyi - yj[t]; ry = __builtin_fmaf(-rintf(ry * INV_BOX), BOXL, ry);
                float rz = czi - zj[t]; rz = __builtin_fmaf(-rintf(rz * INV_BOX), BOXL, rz);
                float r2  = __builtin_fmaf(rx, rx, __builtin_fmaf(ry, ry, rz * rz));
                float r2c = fmaxf(r2, 1.0f);               // clamp(min=1) on d^2
                float id  = __builtin_amdgcn_rsqf(r2c);    // 1/d
                float d   = r2c * id;                      // d
                float id2 = id * id;
                float id6 = id2 * id2 * id2;
                float id8 = id6 * id2;
                float rrho = __builtin_amdgcn_rcpf(rh[t]);
                float E  = __builtin_amdgcn_exp2f((sg[t] - d) * rrho * LN2_INV);
                float AE = a[t] * E;
                float pe = __builtin_fmaf(dd[t], id8, __builtin_fmaf(-cc[t], id6, AE));

                bool valid = (r2c < CUT2) && (iloc != (j & 511));
                float w = valid ? w0 : 0.0f;
                e = __builtin_fmaf(w, pe, e);

                // dpe/dd = -(A/rho)E + 6C/d^7 - 8D/d^9 ; grad zero under clamp
                float dpe = __builtin_fmaf(6.0f * cc[t], id6 * id,
                             __builtin_fmaf(-8.0f * dd[t], id8 * id, -AE * rrho));
                float s = (r2 > 1.0f) ? (w * dpe * id) : 0.0f;  // F = +grad (ref quirk)
                gx[t] = s * rx;
                gy[t] = s * ry;
                gz[t] = s * rz;
            }

            // two WMMA steps cover the 8 columns exactly once:
            //  step0: kh=0 -> jc+0,1 ; kh=1 -> jc+4,5
            //  step1: kh=0 -> jc+2,3 ; kh=1 -> jc+6,7
            v2f a0x = {gx[0], gx[1]}, a1x = {gx[2], gx[3]};
            v2f a0y = {gy[0], gy[1]}, a1y = {gy[2], gy[3]};
            v2f a0z = {gz[0], gz[1]}, a1z = {gz[2], gz[3]};
            accX = __builtin_amdgcn_wmma_f32_16x16x4_f32(false, a0x, false, ones,
                                                         (short)0, accX, false, false);
            accY = __builtin_amdgcn_wmma_f32_16x16x4_f32(false, a0y, false, ones,
                                                         (short)0, accY, false, false);
            accZ = __builtin_amdgcn_wmma_f32_16x16x4_f32(false, a0z, false, ones,
                                                         (short)0, accZ, false, false);
            accX = __builtin_amdgcn_wmma_f32_16x16x4_f32(false, a1x, false, ones,
                                                         (short)0, accX, false, false);
            accY = __builtin_amdgcn_wmma_f32_16x16x4_f32(false, a1y, false, ones,
                                                         (short)0, accY, false, false);
            accZ = __builtin_amdgcn_wmma_f32_16x16x4_f32(false, a1z, false, ones,
                                                         (short)0, accZ, false, false);
        }
    }

    // energy: wave reduction, one atomic per wave (x0.5 for ordered-pair sum)
    for (int off = 16; off; off >>= 1) e += __shfl_xor(e, off, 32);
    if (lane == 0) atomicAdd(out, 0.5f * e);

    // forces: every column of D equals the row total; read column N=0.
    // C/D layout: VGPR r -> M=r (lanes 0-15), M=r+8 (lanes 16-31); N = lane%16.
    if (lane == 0 || lane == 16) {
        const int rbase = i0 + (lane ? 8 : 0);
#pragma unroll
        for (int r = 0; r < 8; ++r) {
            float* f = out + 1 + (size_t)(rbase + r) * 3;
            atomicAdd(f + 0, accX[r]);
            atomicAdd(f + 1, accY[r]);
            atomicAdd(f + 2, accZ[r]);
        }
    }
}

extern "C" void kernel_launch(void* const* d_in, const int* in_sizes, int n_in,
                              void* d_out, int out_size, void* d_ws, size_t ws_size,
                              hipStream_t stream) {
    const float* coords = (const float*)d_in[0];
    // d_in[1] = q : computed-but-unused in the reference energy
    const float* A   = (const float*)d_in[2];
    const float* C   = (const float*)d_in[3];
    const float* D   = (const float*)d_in[4];
    const float* rho = (const float*)d_in[5];
    const float* sig = (const float*)d_in[6];
    float* out = (float*)d_out;

    const int n = out_size > 0 ? out_size : OUT_ELEMS;
    bmh_zero<<<(n + THREADS - 1) / THREADS, THREADS, 0, stream>>>(out, n);
    bmh_kernel<<<NATOMS / ROWS_PER_WAVE, THREADS, 0, stream>>>(coords, A, C, D,
                                                               rho, sig, out);
}